// RTL_84482006712835
// MI455X (gfx1250) — compile-verified
//
#include <hip/hip_runtime.h>
#include <stdint.h>

// CDNA5 / gfx1250 lattice interpolation via V_WMMA_F32_16X16X4_F32.
// out[b,l] = sum_v ( prod_d s_d(bit_d(v)) ) * kernel[l,v],  v in [0,256)
// Factored: v = vh*16+vl ; w[v] = wh[vh]*wl[vl]
//   T[b,vh]  = sum_vl wl[b,vl] * K[l, vh*16+vl]   <- WMMA f32 16x16x(4x4)
//   out[b,l] = sum_vh wh[b,vh] * T[b,vh]          <- 16 VALU FMAs
//
// A-operand per lane only needs the 8 wl values whose vl bit1 == half-wave
// bit (vl bit1 <-> dim 6): fold that into one cndmask on s6, build all
// weights with constant-index scalars (no dynamic-index select trees).

typedef float v2f __attribute__((ext_vector_type(2)));
typedef float v8f __attribute__((ext_vector_type(8)));

typedef __attribute__((address_space(1))) int gas_int;  // global
typedef __attribute__((address_space(3))) int las_int;  // LDS

constexpr int B_TOT = 2048;
constexpr int NFEAT = 128;
constexpr int NLAT  = 512;
constexpr int NVERT = 256;
constexpr int KPAD  = 18;   // 16 + 2 pad: 8B-aligned float2 loads, no bank conflicts

__global__ __launch_bounds__(256) void lattice_wmma_kernel(
    const float* __restrict__ x,      // (B, NFEAT)
    const int*   __restrict__ idx,    // (NLAT, 8)
    const float* __restrict__ kern,   // (NLAT, NVERT)
    float*       __restrict__ out)    // (B, NLAT)
{
  __shared__ float klds[16 * KPAD];          // kernel row, [vh][vl] padded
  __shared__ float tlds[8 * 16 * KPAD];      // per-wave T tiles, [wave][batch][vh]

  const int tid = threadIdx.x;               // 0..255 (8 waves of 32)
  const int ly  = blockIdx.y;                // lattice
  const int bx  = blockIdx.x;                // batch tile of 128

  // ---- Stage this lattice's 256-float kernel row into padded LDS ----------
  {
    const int v = tid;                       // 0..255
    float* dst = &klds[(v >> 4) * KPAD + (v & 15)];
    const float* src = &kern[ly * NVERT + v];
#if __has_builtin(__builtin_amdgcn_global_load_async_to_lds_b32) && \
    __has_builtin(__builtin_amdgcn_s_wait_asynccnt)
    // Async DMA path: global -> LDS without VGPR round-trip (ASYNCcnt).
    __builtin_amdgcn_global_load_async_to_lds_b32(
        (gas_int*)(uintptr_t)src,
        (las_int*)(uintptr_t)dst,
        /*offset=*/0, /*cpol=*/0);
    __builtin_amdgcn_s_wait_asynccnt(0);
#else
    *dst = *src;
#endif
  }
  __syncthreads();

  const int  lane = tid & 31;
  const int  wave = tid >> 5;
  const int  m    = lane & 15;               // batch-in-tile (M) / N column
  const bool hiL  = (lane & 16) != 0;        // upper half-wave
  const int  b    = bx * 128 + wave * 16 + m;

  // ---- Gather + clamp the 8 coordinates (idx address block-uniform) -------
  float xi[8];
#pragma unroll
  for (int d = 0; d < 8; ++d) {
    const int f = idx[ly * 8 + d];
    const float t = x[b * NFEAT + f];
    xi[d] = fminf(fmaxf(t, 0.0f), 1.0f);
  }

  // ---- wh: 16 weights from dims 0..3 (v bits 7..4), constant indices only -
  float wh[16];
  wh[0] = 1.0f;
#pragma unroll
  for (int d = 0; d < 4; ++d) {
#pragma unroll
    for (int j = (1 << d) - 1; j >= 0; --j) {
      const float wv = wh[j];
      wh[2 * j + 1] = wv * xi[d];
      wh[2 * j + 0] = wv * (1.0f - xi[d]);
    }
  }

  // ---- A-operand weights: only the 8 wl values this lane needs ------------
  // vl bits: b3<->d4, b2<->d5, b1<->d6(==hiL), b0<->d7.  chunk c = vl>>2.
  const float c4 = 1.0f - xi[4], c5 = 1.0f - xi[5], c7 = 1.0f - xi[7];
  const float s6 = hiL ? xi[6] : (1.0f - xi[6]);    // single v_cndmask
  float q[4];
  q[0] = (c4 * c5) * s6;
  q[1] = (c4 * xi[5]) * s6;
  q[2] = (xi[4] * c5) * s6;
  q[3] = (xi[4] * xi[5]) * s6;

  // ---- T = wl (16x16) @ K_l^T (16x16) via 4 chained K=4 fp32 WMMAs --------
  v8f acc = {0.f, 0.f, 0.f, 0.f, 0.f, 0.f, 0.f, 0.f};
  const int kbase = m * KPAD + (hiL ? 2 : 0);
#pragma unroll
  for (int c = 0; c < 4; ++c) {
    v2f a;
    a.x = q[c] * c7;      // k even: d7 bit = 0
    a.y = q[c] * xi[7];   // k odd : d7 bit = 1
    const v2f bb = *(const v2f*)&klds[kbase + 4 * c];
    acc = __builtin_amdgcn_wmma_f32_16x16x4_f32(
        /*neg_a=*/false, a, /*neg_b=*/false, bb,
        /*c_mod=*/(short)0, acc, /*reuse_a=*/false, /*reuse_b=*/false);
  }

  // ---- Scatter D tile: lane holds D[M = r + 8*hiL, N = m] in acc[r] -------
  float* tw = tlds + wave * 16 * KPAD;
#pragma unroll
  for (int r = 0; r < 8; ++r) {
    tw[(r + (hiL ? 8 : 0)) * KPAD + m] = acc[r];
  }
  __syncthreads();

  // ---- Final contraction over vh: each lo lane owns one batch row ---------
  if (lane < 16) {
    float s = 0.0f;
#pragma unroll
    for (int j = 0; j < 16; ++j) {
      s = fmaf(wh[j], tw[m * KPAD + j], s);
    }
    out[b * NLAT + ly] = s;
  }
}

extern "C" void kernel_launch(void* const* d_in, const int* in_sizes, int n_in,
                              void* d_out, int out_size, void* d_ws, size_t ws_size,
                              hipStream_t stream) {
  (void)in_sizes; (void)n_in; (void)d_ws; (void)ws_size; (void)out_size;
  const float* x    = (const float*)d_in[0];
  const int*   idx  = (const int*)  d_in[1];
  const float* kern = (const float*)d_in[2];
  float*       out  = (float*)d_out;

  dim3 grid(B_TOT / 128, NLAT);   // 16 batch tiles x 512 lattices
  lattice_wmma_kernel<<<grid, 256, 0, stream>>>(x, idx, kern, out);
}